// Upsample_6554120094013
// MI455X (gfx1250) — compile-verified
//
#include <hip/hip_runtime.h>
#include <hip/hip_bf16.h>
#include <cfloat>

// ---------------------------------------------------------------------------
// Upsample-by-NN: out = (concat(coords, dropped), concat(values, values[:,nn]))
// nn_idx[i] = argmin_j || (dropped[i]-SHIFT) - coords[j] ||
// N = 8192, d = 2, B = 2, C = 64, SCALE = 2 -> N_new = N
// ---------------------------------------------------------------------------

#define TILE    2048          // coords per LDS tile (16 KB as float2)
#define QPW     32            // queries per workgroup (one per lane)
#define NSLICE  8             // waves per workgroup; each wave owns a slice
#define SLICE_LEN (TILE / NSLICE)   // 256 coords per wave per tile

// CDNA5 async global->LDS path (ASYNCcnt). Guarded so the kernel compiles
// even if the toolchain doesn't expose the builtin; fallback is plain
// VGPR-through LDS stores.
#if defined(__AMDGCN__) && __has_builtin(__builtin_amdgcn_global_load_async_to_lds_b128)
#define USE_ASYNC_LDS 1
#else
#define USE_ASYNC_LDS 0
#endif

#if USE_ASYNC_LDS
typedef int v4i __attribute__((ext_vector_type(4)));
typedef v4i __attribute__((address_space(1))) g_v4i;   // global
typedef v4i __attribute__((address_space(3))) l_v4i;   // LDS
#endif

__device__ __forceinline__ void tile_load(float2* __restrict__ dst,
                                          const float2* __restrict__ src,
                                          int tid)
{
    // TILE float2 = 16 KB = 1024 x 16B chunks; 256 threads x 4 chunks.
    const float4* __restrict__ g4 = (const float4*)src;
    float4* __restrict__ s4 = (float4*)dst;
#if USE_ASYNC_LDS
    #pragma unroll
    for (int k = 0; k < 4; ++k) {
        int e = tid + k * 256;
        __builtin_amdgcn_global_load_async_to_lds_b128(
            (g_v4i*)(g4 + e),      // global src (C-style cast: const + AS1)
            (l_v4i*)(s4 + e),      // LDS dst (AS3)
            0, 0);
    }
#else
    #pragma unroll
    for (int k = 0; k < 4; ++k) {
        int e = tid + k * 256;
        s4[e] = g4[e];
    }
#endif
}

__device__ __forceinline__ void wait_async_lds()
{
#if USE_ASYNC_LDS
#if __has_builtin(__builtin_amdgcn_s_wait_asynccnt)
    __builtin_amdgcn_s_wait_asynccnt(0);
#else
    asm volatile("s_wait_asynccnt 0" ::: "memory");
#endif
#endif
}

__global__ __launch_bounds__(256)
void nn_kernel(const float2* __restrict__ coords,
               const float*  __restrict__ dropped,
               int* __restrict__ nn_idx,
               int N)
{
    __shared__ alignas(16) float2 sc[2][TILE];     // double-buffered coord tiles
    __shared__ float rbest[NSLICE * QPW];
    __shared__ int   ridx [NSLICE * QPW];

    const int tid = threadIdx.x;
    const int s   = tid >> 5;        // wave id == coord slice
    const int ql  = tid & 31;        // lane  == query within workgroup
    const int q   = blockIdx.x * QPW + ql;

    const float nx = dropped[2 * q]     - 0.0005f;  // SHIFT = SPACING/2
    const float ny = dropped[2 * q + 1] - 0.0005f;

    // Kick off tile 0 before entering the loop.
    tile_load(sc[0], coords, tid);

    float best = FLT_MAX;
    int   bidx = 0x7fffffff;

    const int ntiles = N / TILE;
    for (int t = 0; t < ntiles; ++t) {
        wait_async_lds();        // our own async loads for tile t done
        __syncthreads();         // everyone's loads visible

        if (t + 1 < ntiles)      // prefetch next tile into the other buffer
            tile_load(sc[(t + 1) & 1], coords + (t + 1) * TILE, tid);

        const float2* __restrict__ sl = sc[t & 1] + s * SLICE_LEN;
        const int base = t * TILE + s * SLICE_LEN;
        #pragma unroll 8
        for (int j = 0; j < SLICE_LEN; ++j) {
            float2 c = sl[j];                 // LDS broadcast: same addr all lanes
            float dx = nx - c.x;
            float dy = ny - c.y;
            float d  = dx * dx + dy * dy;
            if (d < best) { best = d; bidx = base + j; }   // ascending j: first-min kept
        }
        __syncthreads();         // all reads of buffer (t&1) done before reuse
    }

    rbest[tid] = best;
    ridx [tid] = bidx;
    __syncthreads();

    // One lane per query merges the 8 slice partials, with an index tie-break
    // to reproduce jnp.argmin first-occurrence semantics exactly.
    if (tid < QPW) {
        float b  = FLT_MAX;
        int   bi = 0x7fffffff;
        #pragma unroll
        for (int k = 0; k < NSLICE; ++k) {
            float d = rbest[k * QPW + tid];
            int   i = ridx [k * QPW + tid];
            if (d < b || (d == b && i < bi)) { b = d; bi = i; }
        }
        nn_idx[blockIdx.x * QPW + tid] = bi;
    }
}

__global__ __launch_bounds__(256)
void gather_kernel(const float4* __restrict__ vals4,
                   const int*    __restrict__ nn_idx,
                   float4*       __restrict__ outv4,
                   int N, int C4, int B)
{
    int t = blockIdx.x * blockDim.x + threadIdx.x;
    int total = B * N * C4;
    if (t >= total) return;
    int c   = t % C4;
    int rem = t / C4;
    int q   = rem % N;
    int b   = rem / N;
    int j   = nn_idx[q];
    // out row (N + q) of the 2N-row output for batch b
    outv4[(size_t)b * (2 * (size_t)N * C4) + (size_t)(N + q) * C4 + c] =
        vals4[(size_t)b * ((size_t)N * C4) + (size_t)j * C4 + c];
}

extern "C" void kernel_launch(void* const* d_in, const int* in_sizes, int n_in,
                              void* d_out, int out_size, void* d_ws, size_t ws_size,
                              hipStream_t stream)
{
    const float* coords  = (const float*)d_in[0];   // (N, 2) f32
    const float* dropped = (const float*)d_in[1];   // (N, 2) f32
    const float* values  = (const float*)d_in[2];   // (B, N, C) f32
    float* out = (float*)d_out;

    const int N    = in_sizes[0] / 2;               // 8192
    const int C    = 64;
    const int B    = in_sizes[2] / (N * C);         // 2
    const int Nnew = N;                             // SCALE = 2

    // ---- all_coords = concat(coords, dropped[:Nnew]) -------------------
    (void)hipMemcpyAsync(out,         coords,  (size_t)N    * 2 * sizeof(float),
                         hipMemcpyDeviceToDevice, stream);
    (void)hipMemcpyAsync(out + N * 2, dropped, (size_t)Nnew * 2 * sizeof(float),
                         hipMemcpyDeviceToDevice, stream);

    // ---- out_values[:, :N] = values ------------------------------------
    float* outv = out + (size_t)(N + Nnew) * 2;
    for (int b = 0; b < B; ++b) {
        (void)hipMemcpyAsync(outv + (size_t)b * (size_t)(N + Nnew) * C,
                             values + (size_t)b * (size_t)N * C,
                             (size_t)N * C * sizeof(float),
                             hipMemcpyDeviceToDevice, stream);
    }

    // ---- nn_idx --------------------------------------------------------
    int* nn = (int*)d_ws;
    nn_kernel<<<Nnew / QPW, 256, 0, stream>>>((const float2*)coords, dropped, nn, N);

    // ---- out_values[:, N:] = values[:, nn_idx] -------------------------
    const int C4 = C / 4;
    const int total = B * Nnew * C4;
    gather_kernel<<<(total + 255) / 256, 256, 0, stream>>>(
        (const float4*)values, nn, (float4*)outv, N, C4, B);
}